// GraphAutoEncoder_34102040330339
// MI455X (gfx1250) — compile-verified
//
#include <hip/hip_runtime.h>
#include <math.h>

typedef __attribute__((ext_vector_type(2))) float v2f;
typedef __attribute__((ext_vector_type(8))) float v8f;

enum { ACT_NONE = 0, ACT_RELU = 1, ACT_SIGMOID = 2 };

// ---------------------------------------------------------------------------
// Y[nrows x OUT] = act( X[nrows x IN] * W^T + b ),  W is [OUT x IN] row-major.
// One wave32 computes one 16x16 output tile with V_WMMA_F32_16X16X4_F32.
// blockDim.x = (OUT/16)*32 : wave w handles n-tile w; blockIdx.x = m-tile.
//
// f32 WMMA VGPR layout (ISA 7.12.2):
//   A (16x4): lanes 0-15 -> M=lane, VGPR0/1 = K=0/1; lanes 16-31 -> K=2/3.
//   B (4x16): lanes 0-15 -> N=lane, VGPR0/1 = K=0/1; lanes 16-31 -> K=2/3.
//   C/D     : VGPR r, lanes 0-15 -> M=r, N=lane; lanes 16-31 -> M=8+r.
// ---------------------------------------------------------------------------
template <int IN, int OUT, int ACT, bool BIASED>
__global__ __launch_bounds__(512) void linear_wmma(
    const float* __restrict__ X, const float* __restrict__ W,
    const float* __restrict__ bias, float* __restrict__ Y, int nrows) {
  const int lane = threadIdx.x & 31;
  const int wave = threadIdx.x >> 5;      // n-tile within block
  const int m0 = blockIdx.x * 16;         // m-tile
  const int n0 = wave * 16;
  if (m0 + 16 > nrows) return;            // block-uniform (nrows % 16 == 0 here)

  const int half = lane >> 4;             // 0: K pair {0,1}, 1: K pair {2,3}
  const int l = lane & 15;

  const float* __restrict__ arow = X + (size_t)(m0 + l) * IN;
  const float* __restrict__ brow = W + (size_t)(n0 + l) * IN;

  v8f acc = {};
#pragma unroll 8
  for (int k = 0; k < IN; k += 4) {
    v2f a, b;
    a.x = arow[k + 2 * half];
    a.y = arow[k + 2 * half + 1];
    b.x = brow[k + 2 * half];
    b.y = brow[k + 2 * half + 1];
    acc = __builtin_amdgcn_wmma_f32_16x16x4_f32(false, a, false, b, (short)0,
                                                acc, false, false);
  }

  const float bn = BIASED ? bias[n0 + l] : 0.0f;
#pragma unroll
  for (int r = 0; r < 8; ++r) {
    const int m = m0 + r + 8 * half;
    float v = acc[r] + bn;
    if (ACT == ACT_RELU) v = fmaxf(v, 0.0f);
    if (ACT == ACT_SIGMOID) v = 1.0f / (1.0f + expf(-v));
    Y[(size_t)m * OUT + n0 + l] = v;
  }
}

// ---------------------------------------------------------------------------
__global__ void zero_f32(float* __restrict__ p, long long n) {
  long long i = (long long)blockIdx.x * blockDim.x + threadIdx.x;
  if (i < n) p[i] = 0.0f;
}

// deg[dst] += w over E real edges + N self-loops (w=1)
__global__ void degree_kernel(const int* __restrict__ dst,
                              const float* __restrict__ w, int E, int n,
                              float* __restrict__ deg) {
  int i = blockIdx.x * blockDim.x + threadIdx.x;
  if (i >= E + n) return;
  int d;
  float wt;
  if (i < E) {
    d = dst[i];
    wt = w[i];
  } else {
    d = i - E;
    wt = 1.0f;
  }
  atomicAdd(&deg[d], wt);
}

__global__ void dinv_kernel(float* __restrict__ deg, int n) {
  int i = blockIdx.x * blockDim.x + threadIdx.x;
  if (i >= n) return;
  float dg = deg[i];
  deg[i] = dg > 0.0f ? rsqrtf(fmaxf(dg, 1e-12f)) : 0.0f;
}

// acc[dst][f] += tmp[src][f] * dinv[src]*w*dinv[dst];  one thread per (edge,f)
__global__ void scatter64_kernel(const float* __restrict__ tmp,
                                 const int* __restrict__ src,
                                 const int* __restrict__ dst,
                                 const float* __restrict__ w,
                                 const float* __restrict__ dinv,
                                 float* __restrict__ acc, int E, int n) {
  long long idx = (long long)blockIdx.x * blockDim.x + threadIdx.x;
  int e = (int)(idx >> 6);
  int f = (int)(idx & 63);
  if (e >= E + n) return;
  int s, d;
  float wt;
  if (e < E) {
    s = src[e];
    d = dst[e];
    wt = w[e];
  } else {
    s = d = e - E;
    wt = 1.0f;
  }
  float norm = dinv[s] * wt * dinv[d];
  atomicAdd(&acc[(size_t)d * 64 + f], tmp[(size_t)s * 64 + f] * norm);
}

__global__ void bias_relu64(const float* __restrict__ acc,
                            const float* __restrict__ b, float* __restrict__ out,
                            long long n64) {
  long long i = (long long)blockIdx.x * blockDim.x + threadIdx.x;
  if (i >= n64) return;
  out[i] = fmaxf(acc[i] + b[i & 63], 0.0f);
}

// ---------------------------------------------------------------------------
extern "C" void kernel_launch(void* const* d_in, const int* in_sizes, int n_in,
                              void* d_out, int out_size, void* d_ws,
                              size_t ws_size, hipStream_t stream) {
  const float* x = (const float*)d_in[0];
  const int* ei = (const int*)d_in[1];
  const float* ew = (const float*)d_in[2];
  const float* enc1_w = (const float*)d_in[3];
  const float* enc1_b = (const float*)d_in[4];
  const float* enc2_w = (const float*)d_in[5];
  const float* enc2_b = (const float*)d_in[6];
  const float* gcn1_w = (const float*)d_in[7];
  const float* gcn1_b = (const float*)d_in[8];
  const float* gcn2_w = (const float*)d_in[9];
  const float* gcn2_b = (const float*)d_in[10];
  const float* dec1_w = (const float*)d_in[11];
  const float* dec1_b = (const float*)d_in[12];
  const float* dec2_w = (const float*)d_in[13];
  const float* dec2_b = (const float*)d_in[14];

  const int N = in_sizes[0] / 256;   // 50000
  const int E = in_sizes[2];         // 1600000
  const int* src = ei;
  const int* dst = ei + E;

  // workspace layout (floats)
  float* ws = (float*)d_ws;
  float* dinv = ws;                            // N
  float* A = dinv + N;                         // N*128  (h1 / h5)
  float* B = A + (size_t)N * 128;              // N*64   (h2 / h3 / h4)
  float* C = B + (size_t)N * 64;               // N*64   (tmp = h*W^T)
  float* D = C + (size_t)N * 64;               // N*64   (scatter accumulator)

  const int mt = N / 16;                       // 3125 m-tiles
  const long long n64 = (long long)N * 64;
  const long long scatN = (long long)(E + N) * 64;
  const int scatB = (int)((scatN + 255) / 256);

  // --- degree / normalization -------------------------------------------
  zero_f32<<<(N + 255) / 256, 256, 0, stream>>>(dinv, N);
  degree_kernel<<<(E + N + 255) / 256, 256, 0, stream>>>(dst, ew, E, N, dinv);
  dinv_kernel<<<(N + 255) / 256, 256, 0, stream>>>(dinv, N);

  // --- encoder ----------------------------------------------------------
  linear_wmma<256, 128, ACT_RELU, true>
      <<<mt, 8 * 32, 0, stream>>>(x, enc1_w, enc1_b, A, N);
  linear_wmma<128, 64, ACT_RELU, true>
      <<<mt, 4 * 32, 0, stream>>>(A, enc2_w, enc2_b, B, N);

  // --- GCN layer 1 ------------------------------------------------------
  linear_wmma<64, 64, ACT_NONE, false>
      <<<mt, 4 * 32, 0, stream>>>(B, gcn1_w, nullptr, C, N);
  zero_f32<<<(int)((n64 + 255) / 256), 256, 0, stream>>>(D, n64);
  scatter64_kernel<<<scatB, 256, 0, stream>>>(C, src, dst, ew, dinv, D, E, N);
  bias_relu64<<<(int)((n64 + 255) / 256), 256, 0, stream>>>(D, gcn1_b, B, n64);

  // --- GCN layer 2 ------------------------------------------------------
  linear_wmma<64, 64, ACT_NONE, false>
      <<<mt, 4 * 32, 0, stream>>>(B, gcn2_w, nullptr, C, N);
  zero_f32<<<(int)((n64 + 255) / 256), 256, 0, stream>>>(D, n64);
  scatter64_kernel<<<scatB, 256, 0, stream>>>(C, src, dst, ew, dinv, D, E, N);
  bias_relu64<<<(int)((n64 + 255) / 256), 256, 0, stream>>>(D, gcn2_b, B, n64);

  // --- decoder ----------------------------------------------------------
  linear_wmma<64, 128, ACT_RELU, true>
      <<<mt, 8 * 32, 0, stream>>>(B, dec1_w, dec1_b, A, N);
  linear_wmma<128, 256, ACT_SIGMOID, true>
      <<<mt, 16 * 32, 0, stream>>>(A, dec2_w, dec2_b, (float*)d_out, N);
}